// ST_Rec_Module_23278722744415
// MI455X (gfx1250) — compile-verified
//
#include <hip/hip_runtime.h>
#include <hip/hip_bf16.h>

#define BB 8
#define CC 1024
#define KK 8
#define PP 128
#define RR 128

typedef __attribute__((ext_vector_type(2))) float v2f;
typedef __attribute__((ext_vector_type(8))) float v8f;

// ---------------------------------------------------------------------------
// Kernel 1: projections  x_q = q @ Wq, x_k = k @ Wk   (fp32 WMMA 16x16x4)
//   grid = (BB*CC/16, 2), block = 256 (8 waves; wave w -> column tile w of R)
// ---------------------------------------------------------------------------
__global__ void st_proj_kernel(const float* __restrict__ q,
                               const float* __restrict__ k,
                               const float* __restrict__ Wq,
                               const float* __restrict__ Wk,
                               float* __restrict__ xq,
                               float* __restrict__ xk) {
  const int sel  = blockIdx.y;                 // 0 -> q/Wq, 1 -> k/Wk
  const float* __restrict__ src = sel ? k  : q;
  const float* __restrict__ W   = sel ? Wk : Wq;
  float* __restrict__ dst       = sel ? xk : xq;

  const int lane = threadIdx.x & 31;
  const int wv   = threadIdx.x >> 5;           // 0..7 -> column tile
  const int half = lane >> 4;                  // 0 or 1
  const int lid  = lane & 15;
  const int kslot = 2 * half;                  // K sub-slot: {0,1} or {2,3}

  const int rowBase = blockIdx.x * 16;         // rows in flattened (B*C, P)
  const int colBase = wv * 16;                 // cols in R

  const float* arow = src + (size_t)(rowBase + lid) * PP;

  v8f acc = {0.f, 0.f, 0.f, 0.f, 0.f, 0.f, 0.f, 0.f};
  #pragma unroll 4
  for (int kb = 0; kb < PP; kb += 4) {
    v2f a, b;
    a[0] = arow[kb + kslot + 0];
    a[1] = arow[kb + kslot + 1];
    b[0] = W[(size_t)(kb + kslot + 0) * RR + colBase + lid];
    b[1] = W[(size_t)(kb + kslot + 1) * RR + colBase + lid];
    acc = __builtin_amdgcn_wmma_f32_16x16x4_f32(false, a, false, b,
                                                (short)0, acc, false, false);
  }

  #pragma unroll
  for (int r = 0; r < 8; ++r) {
    dst[(size_t)(rowBase + r + 8 * half) * RR + colBase + lid] = acc[r];
  }
}

// ---------------------------------------------------------------------------
// Kernel 2: fused logits (WMMA) + top-8 + gather of v rows.
//   grid = (CC/16, BB), block = 256 (8 waves; wave w -> keys [w*128, w*128+128))
//   LDS: 16x1024 fp32 logits strip + top-k scratch (~82.5 KB of 320 KB/WGP)
// ---------------------------------------------------------------------------
__global__ void st_topk_gather_kernel(const float* __restrict__ xq,
                                      const float* __restrict__ xk,
                                      const float* __restrict__ v,
                                      float* __restrict__ out) {
  __shared__ float s_logits[16 * CC];          // 64 KB
  __shared__ float s_cval[16 * 16 * KK];       // 8 KB  candidate values
  __shared__ int   s_cidx[16 * 16 * KK];       // 8 KB  candidate indices
  __shared__ int   s_topk[16 * KK];            // final indices per row

  const int b     = blockIdx.y;
  const int qbase = blockIdx.x * 16;

  const int tid  = threadIdx.x;
  const int lane = tid & 31;
  const int wv   = tid >> 5;                   // 0..7
  const int half = lane >> 4;
  const int lid  = lane & 15;
  const int kslot = 2 * half;

  // ---- phase 1: logits[16 x 1024] via fp32 WMMA -------------------------
  const float* qrow = xq + (size_t)(b * CC + qbase + lid) * RR;

  v8f acc[8];
  const v8f vzero = {0.f, 0.f, 0.f, 0.f, 0.f, 0.f, 0.f, 0.f};
  #pragma unroll
  for (int kt = 0; kt < 8; ++kt) acc[kt] = vzero;

  for (int kb = 0; kb < RR; kb += 4) {
    v2f a;
    a[0] = qrow[kb + kslot + 0];
    a[1] = qrow[kb + kslot + 1];
    #pragma unroll
    for (int kt = 0; kt < 8; ++kt) {
      const float* krow = xk + (size_t)(b * CC + wv * 128 + kt * 16 + lid) * RR;
      v2f bf;
      bf[0] = krow[kb + kslot + 0];
      bf[1] = krow[kb + kslot + 1];
      acc[kt] = __builtin_amdgcn_wmma_f32_16x16x4_f32(false, a, false, bf,
                                                      (short)0, acc[kt],
                                                      false, false);
    }
  }

  #pragma unroll
  for (int kt = 0; kt < 8; ++kt) {
    #pragma unroll
    for (int r = 0; r < 8; ++r) {
      s_logits[(r + 8 * half) * CC + wv * 128 + kt * 16 + lid] = acc[kt][r];
    }
  }
  __syncthreads();

  // ---- phase 2: top-8 per row (JAX tie-break: lower index on equal) -----
  {
    const int row = tid >> 4;                  // 0..15
    const int sub = tid & 15;                  // 0..15 -> 64-key stripe
    float bv[KK];
    int   bi[KK];
    #pragma unroll
    for (int t = 0; t < KK; ++t) { bv[t] = -__builtin_inff(); bi[t] = 0x7fffffff; }

    const float* lrow = s_logits + row * CC;
    for (int i = 0; i < 64; ++i) {
      const int   idx = sub * 64 + i;
      const float val = lrow[idx];
      #pragma unroll
      for (int t = 0; t < KK; ++t) {
        if (val > bv[t] || (val == bv[t] && idx < bi[t])) {
          #pragma unroll
          for (int s = KK - 1; s > 0; --s) {
            if (s > t) { bv[s] = bv[s - 1]; bi[s] = bi[s - 1]; }
          }
          bv[t] = val; bi[t] = idx;
          break;
        }
      }
    }
    #pragma unroll
    for (int t = 0; t < KK; ++t) {
      s_cval[(row * 16 + sub) * KK + t] = bv[t];
      s_cidx[(row * 16 + sub) * KK + t] = bi[t];
    }
  }
  __syncthreads();

  if (tid < 16) {
    const int row = tid;
    float bv[KK];
    int   bi[KK];
    #pragma unroll
    for (int t = 0; t < KK; ++t) { bv[t] = -__builtin_inff(); bi[t] = 0x7fffffff; }
    for (int c = 0; c < 16 * KK; ++c) {
      const float val = s_cval[row * 16 * KK + c];
      const int   idx = s_cidx[row * 16 * KK + c];
      #pragma unroll
      for (int t = 0; t < KK; ++t) {
        if (val > bv[t] || (val == bv[t] && idx < bi[t])) {
          #pragma unroll
          for (int s = KK - 1; s > 0; --s) {
            if (s > t) { bv[s] = bv[s - 1]; bi[s] = bi[s - 1]; }
          }
          bv[t] = val; bi[t] = idx;
          break;
        }
      }
    }
    #pragma unroll
    for (int t = 0; t < KK; ++t) s_topk[row * KK + t] = bi[t];
  }
  __syncthreads();

  // ---- phase 3: gather  out[b, qbase+row, t, :] = v[b, idx, :] ----------
  {
    const int pid  = tid >> 1;                 // 0..127 -> (row, t)
    const int hh   = tid & 1;                  // which 64-float half
    const int row  = pid >> 3;
    const int t    = pid & 7;
    const int idx  = s_topk[row * KK + t];

    const float4* srcv = (const float4*)(v + ((size_t)b * CC + idx) * PP + hh * 64);
    float4* dstv = (float4*)(out +
        ((((size_t)b * CC + qbase + row) * KK + t) * PP) + hh * 64);
    #pragma unroll
    for (int i = 0; i < 16; ++i) dstv[i] = srcv[i];
  }
}

extern "C" void kernel_launch(void* const* d_in, const int* in_sizes, int n_in,
                              void* d_out, int out_size, void* d_ws, size_t ws_size,
                              hipStream_t stream) {
  const float* q  = (const float*)d_in[0];
  const float* k  = (const float*)d_in[1];
  const float* v  = (const float*)d_in[2];
  const float* Wq = (const float*)d_in[3];
  const float* Wk = (const float*)d_in[4];
  float* out = (float*)d_out;

  float* xq = (float*)d_ws;                          // B*C*R fp32 = 4 MB
  float* xk = xq + (size_t)BB * CC * RR;             // next 4 MB

  st_proj_kernel<<<dim3(BB * CC / 16, 2), 256, 0, stream>>>(q, k, Wq, Wk, xq, xk);
  st_topk_gather_kernel<<<dim3(CC / 16, BB), 256, 0, stream>>>(xq, xk, v, out);
}